// KSubspaceBaseModel_57509612093853
// MI455X (gfx1250) — compile-verified
//
#include <hip/hip_runtime.h>
#include <hip/hip_bf16.h>

// Problem constants (match reference)
#define R_ 8
#define K_ 10
#define D_ 256
#define DL_ 10
#define B_ 4096
#define KD_ 100          // K_*DL_
#define KDP_ 112         // KD_ padded to 7*16
#define INV_TEMP 10.0f   // 1/0.1

#define XP 260           // LDS pitch for x tile (floats): lanes hit distinct banks
#define ZP 114           // LDS pitch for z tile: distinct banks + 8B-aligned pairs

// Workspace layout (bytes)
#define G_OFF  0u              // 80*100*4 = 32000 B
#define UP_OFF 32768u          // Upad2 : 8*128*112*2*4 = 917504 B
#define UT_OFF 950272u         // UpadT2: 8*56*256*2*4  = 917504 B
// total ws needed: 1,867,776 B

typedef float v2f __attribute__((ext_vector_type(2)));
typedef float v8f __attribute__((ext_vector_type(8)));

// ---------------------------------------------------------------------------
// Prep A: Gram blocks  G[r,k] = Us[r,k]^T Us[r,k]  (10x10 each, 80 blocks)
// ---------------------------------------------------------------------------
__global__ __launch_bounds__(128) void gram_kernel(const float* __restrict__ Us,
                                                   float* __restrict__ G) {
  const int rk = blockIdx.x;        // 0..79
  const int t  = threadIdx.x;       // first 100 active
  if (t >= KD_) return;
  const int d1 = t / DL_, d2 = t % DL_;
  const float* u = Us + (size_t)rk * D_ * DL_;
  float acc = 0.f;
#pragma unroll 8
  for (int Dd = 0; Dd < D_; ++Dd)
    acc = fmaf(u[Dd * DL_ + d1], u[Dd * DL_ + d2], acc);
  G[rk * KD_ + t] = acc;
}

// ---------------------------------------------------------------------------
// Prep B: pack U into zero-padded WMMA-friendly layouts.
//   Ucat_r[Dd][kd] = Us[r, kd/10, Dd, kd%10]  (kd<100), else 0
//   Upad2 [r][Dd>>1][kd][Dd&1]  -> GEMM1 B operand, b64-contiguous K-pairs
//   UpadT2[r][kd>>1][Dd][kd&1]  -> GEMM2 B operand (transposed), same trick
// ---------------------------------------------------------------------------
__global__ __launch_bounds__(KDP_) void pack_kernel(const float* __restrict__ Us,
                                                    float* __restrict__ Upad2,
                                                    float* __restrict__ UpadT2) {
  const int kd = threadIdx.x;       // 0..111
  const int rd = blockIdx.x;        // r*256 + Dd
  const int r  = rd >> 8;
  const int Dd = rd & 255;
  float val = 0.f;
  if (kd < KD_) {
    const int k = kd / DL_, d = kd % DL_;
    val = Us[(((size_t)r * K_ + k) * D_ + Dd) * DL_ + d];
  }
  Upad2 [(((size_t)r * (D_ / 2) + (Dd >> 1)) * KDP_ + kd) * 2 + (Dd & 1)] = val;
  UpadT2[(((size_t)r * (KDP_ / 2) + (kd >> 1)) * D_ + Dd) * 2 + (kd & 1)] = val;
}

// ---------------------------------------------------------------------------
// Fused kernel: one wave32 per (16-row b-tile, r).
//   GEMM1: z = xtile @ Ucat_r            (16x256 @ 256x112, pad cols -> 0)
//   loss_k = 0.5*z_k^T G_k z_k - ||z_k||^2 (+const) -> softmax over k -> c
//   w = c .* z ;  out_tile = w @ Ucat_r^T  (16x100 @ 100x256)
// ---------------------------------------------------------------------------
__global__ __launch_bounds__(32) void ksub_fused(const float* __restrict__ x,
                                                 const float* __restrict__ Upad2,
                                                 const float* __restrict__ UpadT2,
                                                 const float* __restrict__ G,
                                                 float* __restrict__ out) {
  __shared__ float sx[16 * XP];     // x tile 16x256
  __shared__ float sz[16 * ZP];     // z / w tile 16x112
  __shared__ float sg[K_ * KD_];    // Gram blocks for this r

  const int lane = threadIdx.x;     // wave32
  const int r  = blockIdx.y;
  const int b0 = blockIdx.x * 16;
  const int g  = lane >> 4;         // half-wave group (0/1)
  const int n  = lane & 15;

  // ---- cooperative coalesced load: x tile -> LDS ----
  for (int i = lane; i < 16 * 64; i += 32) {
    const int row = i >> 6;
    const int c4  = (i & 63) << 2;
    const float4 v = *(const float4*)(x + (size_t)(b0 + row) * D_ + c4);
    *(float4*)(&sx[row * XP + c4]) = v;
  }
  for (int i = lane; i < K_ * KD_; i += 32) sg[i] = G[r * K_ * KD_ + i];
  __syncthreads();

  // ---- GEMM 1: 7 N-tiles of 16 over kd, K = 256 (64 WMMA steps each) ----
  const float* Up = Upad2 + (size_t)r * (D_ / 2) * KDP_ * 2;
#pragma unroll 1
  for (int t = 0; t < 7; ++t) {
    const int kd = t * 16 + n;                       // this lane's N column
    const float* ub = Up + ((size_t)g * KDP_ + kd) * 2;  // K-pair base
    v8f acc = {};
#pragma unroll
    for (int kb = 0; kb < D_; kb += 4) {
      // A (16x4): M = n, K = kb+2g, kb+2g+1  -> one ds_load_b64
      v2f a = *(const v2f*)(&sx[n * XP + kb + 2 * g]);
      // B (4x16): N = kd, same Ks            -> one global_load_b64, imm offset
      v2f b = *(const v2f*)(&ub[(size_t)(kb >> 1) * KDP_ * 2]);
      acc = __builtin_amdgcn_wmma_f32_16x16x4_f32(false, a, false, b,
                                                  (short)0, acc, false, false);
    }
#pragma unroll
    for (int v = 0; v < 8; ++v)                      // C/D: M = v+8g, N = kd
      sz[(v + 8 * g) * ZP + kd] = acc[v];
  }
  __syncthreads();

  // ---- soft assignment: lane L (<16) owns b-row L ----
  if (lane < 16) {
    float s[K_];
    float m = -3.0e38f;
#pragma unroll 1
    for (int k = 0; k < K_; ++k) {
      float zk[DL_];
#pragma unroll
      for (int d = 0; d < DL_; ++d) zk[d] = sz[lane * ZP + k * DL_ + d];
      float zz = 0.f, q = 0.f;
#pragma unroll
      for (int d1 = 0; d1 < DL_; ++d1) {
        zz = fmaf(zk[d1], zk[d1], zz);
        float t2 = 0.f;
#pragma unroll
        for (int d2 = 0; d2 < DL_; ++d2)
          t2 = fmaf(sg[k * KD_ + d1 * DL_ + d2], zk[d2], t2);
        q = fmaf(zk[d1], t2, q);
      }
      // -loss/T up to a k-independent constant (0.5||x||^2 cancels in softmax)
      s[k] = (zz - 0.5f * q) * INV_TEMP;
      m = fmaxf(m, s[k]);
    }
    float c[K_];
    float sum = 0.f;
#pragma unroll
    for (int k = 0; k < K_; ++k) { c[k] = __expf(s[k] - m); sum += c[k]; }
    const float inv = 1.0f / sum;
#pragma unroll 1
    for (int k = 0; k < K_; ++k) {
      const float ck = c[k] * inv;
#pragma unroll
      for (int d = 0; d < DL_; ++d) sz[lane * ZP + k * DL_ + d] *= ck;
    }
  }
  __syncthreads();

  // ---- GEMM 2: out tile (16x256) = w (16x100) @ Ucat_r^T (100x256) ----
  const float* Ut = UpadT2 + (size_t)r * (KDP_ / 2) * D_ * 2;
  float* outr = out + (size_t)r * B_ * D_ + (size_t)b0 * D_;
#pragma unroll 1
  for (int t = 0; t < 16; ++t) {
    const int Dd = t * 16 + n;                       // this lane's N column
    const float* ubt = Ut + ((size_t)g * D_ + Dd) * 2;
    v8f acc = {};
#pragma unroll
    for (int kb = 0; kb < KD_; kb += 4) {            // K = 100, 25 steps
      // A: M = n, K = kb+2g, kb+2g+1 (8B-aligned: ZP even, kb even)
      v2f a = *(const v2f*)(&sz[n * ZP + kb + 2 * g]);
      // B: N = Dd, same Ks -> one global_load_b64, imm offset
      v2f b = *(const v2f*)(&ubt[(size_t)(kb >> 1) * D_ * 2]);
      acc = __builtin_amdgcn_wmma_f32_16x16x4_f32(false, a, false, b,
                                                  (short)0, acc, false, false);
    }
#pragma unroll
    for (int v = 0; v < 8; ++v)
      outr[(size_t)(v + 8 * g) * D_ + Dd] = acc[v];
  }
}

// ---------------------------------------------------------------------------
extern "C" void kernel_launch(void* const* d_in, const int* in_sizes, int n_in,
                              void* d_out, int out_size, void* d_ws, size_t ws_size,
                              hipStream_t stream) {
  (void)in_sizes; (void)n_in; (void)out_size; (void)ws_size;
  const float* x  = (const float*)d_in[0];   // (B, D) fp32
  const float* Us = (const float*)d_in[1];   // (R, K, D, DLOW) fp32
  float* G      = (float*)((char*)d_ws + G_OFF);
  float* Upad2  = (float*)((char*)d_ws + UP_OFF);
  float* UpadT2 = (float*)((char*)d_ws + UT_OFF);
  float* out = (float*)d_out;                // (R, B, D) fp32

  gram_kernel<<<R_ * K_, 128, 0, stream>>>(Us, G);
  pack_kernel<<<R_ * D_, KDP_, 0, stream>>>(Us, Upad2, UpadT2);
  ksub_fused<<<dim3(B_ / 16, R_), 32, 0, stream>>>(x, Upad2, UpadT2, G, out);
}